// SingleT1FLS_TSK_3418793968081
// MI455X (gfx1250) — compile-verified
//
#include <hip/hip_runtime.h>

// TSK type-1 fuzzy inference, B=65536, R=32, A=16.
// Reformulated as f32 WMMA GEMMs [B,16+const]x[16+const,32]:
//   Q[b,r]  = x2@(-a*log2e) + x@(2ac*log2e) + (-sum(ac^2)*log2e)   (const K-chunk)
//   CH[b,r] = x@C[:,1:] + C[:,0]                                   (const K-chunk)
//   UU      = exp2(Q)        -> single v_exp_f32 per element
//   out[b]  = sum_r UU*CH / sum_r UU
// Weights prepped once into d_ws in WMMA-fragment order (b128 loads in hot path).

#define RULES 32
#define ANT   16
#define LOG2E 1.4426950408889634f

typedef __attribute__((ext_vector_type(2))) float v2f;
typedef __attribute__((ext_vector_type(4))) float v4f;
typedef __attribute__((ext_vector_type(8))) float v8f;

// d_ws float layout. Weight matrices stored in fragment order:
//   idx = ((t*2 + h)*16 + n)*8 + kc*2 + j   for W[k=4*kc+2*h+j, r=t*16+n]
#define WS_WQ   0      // [512]  -a*log2e
#define WS_WL   512    // [512]  +2ac*log2e
#define WS_WC   1024   // [512]  C[r, k+1]
#define WS_DR   1536   // [32]   -sum_k a*c^2 * log2e
#define WS_BIAS 1568   // [32]   C[r, 0]
#define WS_FLOATS 1600

__device__ __forceinline__ v8f wmma4(v2f a, v2f b, v8f c) {
    return __builtin_amdgcn_wmma_f32_16x16x4_f32(false, a, false, b, (short)0, c,
                                                 false, false);
}
__device__ __forceinline__ v2f lo2(v4f v) { return (v2f){v.x, v.y}; }
__device__ __forceinline__ v2f hi2(v4f v) { return (v2f){v.z, v.w}; }

__global__ void tsk_prep_kernel(const float* __restrict__ frb,
                                const float* __restrict__ Cm,
                                float* __restrict__ ws) {
    const int r = threadIdx.x;
    if (r < RULES) {
        const int t = r >> 4, n = r & 15;
        float d = 0.0f;
#pragma unroll 1
        for (int k = 0; k < ANT; ++k) {
            // overlapping windows: sigma = frb[16r+k], center = frb[16r+k+1]
            float s = frb[ANT * r + k];
            float c = frb[ANT * r + k + 1];
            float a = 0.5f / (s * s);            // runs once; full precision
            const int h = (k >> 1) & 1, kc = k >> 2, j = k & 1;
            const int fi = ((t * 2 + h) * 16 + n) * 8 + kc * 2 + j;
            ws[WS_WQ + fi] = -a * LOG2E;
            ws[WS_WL + fi] = 2.0f * a * c * LOG2E;
            ws[WS_WC + fi] = Cm[r * (ANT + 1) + k + 1];
            d += a * c * c;
        }
        ws[WS_DR + r]   = -d * LOG2E;
        ws[WS_BIAS + r] = Cm[r * (ANT + 1)];
    }
}

__global__ __launch_bounds__(256) void tsk_wmma_kernel(
    const float* __restrict__ x,    // [B,16]
    const float* __restrict__ ws,   // prepped weights (fragment order)
    float* __restrict__ out)        // [B]
{
    const int lane = threadIdx.x & 31;
    const int wid  = threadIdx.x >> 5;
    const int tile = blockIdx.x * 8 + wid;   // 16-sample tile per wave
    const int n    = lane & 15;
    const int h    = lane >> 4;

    // ---- A fragments (x, x^2): lane -> M = n; chunk kc VGPR j -> K = 4kc+2h+j
    const float* xrow = x + (size_t)(tile * 16 + n) * ANT;
    v2f xa[4], x2a[4];
#pragma unroll
    for (int kc = 0; kc < 4; ++kc) {
        const int k = kc * 4 + h * 2;
        float x0 = xrow[k];
        float x1 = xrow[k + 1];
        xa[kc]  = (v2f){x0, x1};
        x2a[kc] = (v2f){x0 * x0, x1 * x1};
    }
    // constant K-chunk A fragment: A[m,0]=1, else 0  (folds dr/bias into GEMM)
    const v2f aone = (v2f){h ? 0.0f : 1.0f, 0.0f};

    // vals[0..7] = num[v], vals[8..15] = den[v]
    float vals[16];
#pragma unroll
    for (int i = 0; i < 16; ++i) vals[i] = 0.0f;

#pragma unroll
    for (int t = 0; t < 2; ++t) {            // rule halves: r = t*16 + n
        const int r    = t * 16 + n;
        const int base = ((t * 2 + h) * 16 + n) * 8;
        const v4f* wq4 = (const v4f*)(ws + WS_WQ + base);
        const v4f* wl4 = (const v4f*)(ws + WS_WL + base);
        const v4f* wc4 = (const v4f*)(ws + WS_WC + base);
        v4f wqA = wq4[0], wqB = wq4[1];      // 2 x b128 per matrix
        v4f wlA = wl4[0], wlB = wl4[1];
        v4f wcA = wc4[0], wcB = wc4[1];
        const float drs  = ws[WS_DR + r];
        const float bias = ws[WS_BIAS + r];

        // constant chunk seeds the accumulator chains (C = inline 0)
        v8f q  = wmma4(aone, (v2f){drs, 0.0f},  (v8f){});
        v8f ch = wmma4(aone, (v2f){bias, 0.0f}, (v8f){});
        q  = wmma4(x2a[0], lo2(wqA), q);
        q  = wmma4(xa[0],  lo2(wlA), q);
        ch = wmma4(xa[0],  lo2(wcA), ch);
        q  = wmma4(x2a[1], hi2(wqA), q);
        q  = wmma4(xa[1],  hi2(wlA), q);
        ch = wmma4(xa[1],  hi2(wcA), ch);
        q  = wmma4(x2a[2], lo2(wqB), q);
        q  = wmma4(xa[2],  lo2(wlB), q);
        ch = wmma4(xa[2],  lo2(wcB), ch);
        q  = wmma4(x2a[3], hi2(wqB), q);
        q  = wmma4(xa[3],  hi2(wlB), q);
        ch = wmma4(xa[3],  hi2(wcB), ch);

        // C/D layout: VGPR v, lane -> M = v + 8h, N = n (= rule column)
#pragma unroll
        for (int v = 0; v < 8; ++v) {
            float uu = __builtin_amdgcn_exp2f(q[v]);   // UU = exp2(Q)
            vals[v]     = __builtin_fmaf(uu, ch[v], vals[v]);
            vals[8 + v] += uu;
        }
    }

    // ---- folding reduction over 16 lanes of each half: 15 shuffles total ----
    // final: lane bits l0..l3 -> value idx = l0*8 + l1*4 + l2*2 + l3
#pragma unroll
    for (int s = 0; s < 4; ++s) {
        const int mask  = 1 << s;
        const int nhalf = 8 >> s;
        const bool up   = (lane >> s) & 1;
#pragma unroll
        for (int i = 0; i < nhalf; ++i) {
            float keep = up ? vals[i + nhalf] : vals[i];
            float send = up ? vals[i] : vals[i + nhalf];
            vals[i] = keep + __shfl_xor(send, mask, 32);
        }
    }
    // lane bit0 = (num vs den); remaining bits give sample row v
    float mine    = vals[0];
    float partner = __shfl_xor(mine, 1, 32);
    if ((lane & 1) == 0) {
        const int v = (((lane >> 1) & 1) << 2) | (((lane >> 2) & 1) << 1) | ((lane >> 3) & 1);
        out[tile * 16 + v + 8 * h] = mine / partner;   // 16 lanes, one store
    }
}

extern "C" void kernel_launch(void* const* d_in, const int* in_sizes, int n_in,
                              void* d_out, int out_size, void* d_ws, size_t ws_size,
                              hipStream_t stream) {
    const float* x   = (const float*)d_in[0];  // [B,16] f32
    const float* frb = (const float*)d_in[1];  // [1024] f32
    const float* Cm  = (const float*)d_in[2];  // [32,17] f32
    float* out = (float*)d_out;                // [B] f32
    float* ws  = (float*)d_ws;                 // needs WS_FLOATS*4 = 6400 B

    tsk_prep_kernel<<<1, 64, 0, stream>>>(frb, Cm, ws);

    const int Bsamples = in_sizes[0] / ANT;    // 65536
    const int tiles    = Bsamples / 16;        // one 16-sample tile per wave
    const int blocks   = tiles / 8;            // 8 waves per 256-thread block

    tsk_wmma_kernel<<<blocks, 256, 0, stream>>>(x, ws, out);
}